// MaskedSelfAttention_77000173683527
// MI455X (gfx1250) — compile-verified
//
#include <hip/hip_runtime.h>
#include <hip/hip_bf16.h>
#include <math.h>

// Masked self-attention decode step, algebraically reduced to GEMV form.
// DIMS = 4096, rows = 8192 (8191 prev + 1 new).

#define DIMS   4096
#define NROWS  8192
#define NPREV  8191

typedef __attribute__((ext_vector_type(2))) float v2f;
typedef __attribute__((ext_vector_type(8))) float v8f;

// ---------------------------------------------------------------------------
// y[row] = dot(M[row,:], x) (+ resid[row]).  One wave (32 lanes) per row.
// Per-iteration a wave touches 512 contiguous bytes of M -> full-line reads.
// ---------------------------------------------------------------------------
__global__ __launch_bounds__(256)
void gemv_rows_kernel(const float* __restrict__ M, const float* __restrict__ x,
                      const float* __restrict__ resid, float* __restrict__ y) {
    const int wave = threadIdx.x >> 5;
    const int lane = threadIdx.x & 31;
    const int row  = blockIdx.x * 8 + wave;
    const float4* Mr = (const float4*)(M + (size_t)row * DIMS);
    const float4* x4 = (const float4*)x;
    float acc = 0.f;
    for (int i = lane; i < DIMS / 4; i += 32) {
        float4 a = Mr[i];
        float4 b = x4[i];
        acc += a.x * b.x + a.y * b.y + a.z * b.z + a.w * b.w;
    }
    for (int off = 16; off > 0; off >>= 1) acc += __shfl_xor(acc, off, 32);
    if (lane == 0) y[row] = resid ? acc + resid[row] : acc;
}

// ---------------------------------------------------------------------------
// kq_part[rb][d] = sum_{m in chunk rb} W_K[m][d] * q[m]
// Column-major GEMV: 16 column tiles x 8 row chunks; q chunk staged in LDS.
// ---------------------------------------------------------------------------
__global__ __launch_bounds__(256)
void kq_partial_kernel(const float* __restrict__ WK, const float* __restrict__ q,
                       float* __restrict__ kq_part) {
    __shared__ float qs[512];
    const int cb = blockIdx.x & 15;   // column tile (256 cols)
    const int rb = blockIdx.x >> 4;   // row chunk  (512 rows)
    const int d  = cb * 256 + threadIdx.x;
    for (int i = threadIdx.x; i < 512; i += 256) qs[i] = q[rb * 512 + i];
    __syncthreads();
    const float* col = WK + (size_t)(rb * 512) * DIMS + d;
    float acc = 0.f;
    #pragma unroll 8
    for (int m = 0; m < 512; ++m) acc = fmaf(col[(size_t)m * DIMS], qs[m], acc);
    kq_part[rb * DIMS + d] = acc;
}

// outv[d] = sum_p part[p][d]   (deterministic partial reduction, 16 blocks)
__global__ __launch_bounds__(256)
void reduce_part_kernel(const float* __restrict__ part, float* __restrict__ outv,
                        int nparts) {
    const int d = blockIdx.x * 256 + threadIdx.x;
    float s = 0.f;
    for (int p = 0; p < nparts; ++p) s += part[(size_t)p * DIMS + d];
    outv[d] = s;
}

// ---------------------------------------------------------------------------
// Fused: out_inputs[n,:] = inputs[n,:]  AND  logits[n] = dot(inputs[n,:], kq).
// Carries the dominant 268 MB of HBM traffic; one block per sequence row.
// ---------------------------------------------------------------------------
__global__ __launch_bounds__(256)
void logits_copy_kernel(const float* __restrict__ prev, const float* __restrict__ inp,
                        const float* __restrict__ kq, float* __restrict__ out_inputs,
                        float* __restrict__ logits) {
    const int n = blockIdx.x;
    const float4* src = (const float4*)((n < NPREV) ? (prev + (size_t)n * DIMS) : inp);
    float4*       dst = (float4*)(out_inputs + (size_t)n * DIMS);
    const float4* kq4 = (const float4*)kq;
    float acc = 0.f;
    for (int i = threadIdx.x; i < DIMS / 4; i += 256) {
        float4 v = src[i];
        dst[i] = v;
        float4 k = kq4[i];
        acc += v.x * k.x + v.y * k.y + v.z * k.z + v.w * k.w;
    }
    __shared__ float red[256];
    red[threadIdx.x] = acc;
    __syncthreads();
    for (int s = 128; s > 0; s >>= 1) {
        if (threadIdx.x < s) red[threadIdx.x] += red[threadIdx.x + s];
        __syncthreads();
    }
    if (threadIdx.x == 0) logits[n] = red[0];
}

// Softmax over 8192 logits, single block.
__global__ __launch_bounds__(256)
void softmax_kernel(const float* __restrict__ logits, float* __restrict__ w) {
    __shared__ float red[256];
    float m = -INFINITY;
    for (int i = threadIdx.x; i < NROWS; i += 256) m = fmaxf(m, logits[i]);
    red[threadIdx.x] = m;
    __syncthreads();
    for (int s = 128; s > 0; s >>= 1) {
        if (threadIdx.x < s) red[threadIdx.x] = fmaxf(red[threadIdx.x], red[threadIdx.x + s]);
        __syncthreads();
    }
    const float gmax = red[0];
    __syncthreads();
    float sum = 0.f;
    for (int i = threadIdx.x; i < NROWS; i += 256) {
        float e = expf(logits[i] - gmax);
        w[i] = e;
        sum += e;
    }
    red[threadIdx.x] = sum;
    __syncthreads();
    for (int s = 128; s > 0; s >>= 1) {
        if (threadIdx.x < s) red[threadIdx.x] += red[threadIdx.x + s];
        __syncthreads();
    }
    const float inv = 1.0f / red[0];
    for (int i = threadIdx.x; i < NROWS; i += 256) w[i] *= inv;
}

// ---------------------------------------------------------------------------
// c_part[chunk][d] = sum_{k in chunk} w[k] * inputs[k][d]  via
// V_WMMA_F32_16X16X4_F32.
//
// A (16x4 f32, row M=0 = weights, rows 1..15 = 0):
//   all lanes do one unconditional ds_load_b64 of wl[k + half]
//   (lanes 0-15 -> K0,K1; lanes 16-31 -> K2,K3), then scale by the per-lane
//   constant sel = ((lane&15)==0) ? 1 : 0.  Branch-free: EXEC stays all-1s
//   and the loop body is ds_load + 2 global_load + 2 v_mul + v_wmma.
// B (4x16 f32): VGPR0 lanes0-15 = inputs row k (64 contiguous bytes),
//   lanes16-31 = row k+2; VGPR1 = rows k+1 / k+3.
// Two independent accumulator chains (even/odd K-steps) break the serial
// WMMA->WMMA C dependency; only C row M=0 (element 0, lanes 0-15) is live.
// 4 seq chunks x 256 d-tiles = 1024 waves; weights chunk staged in LDS.
// ---------------------------------------------------------------------------
__global__ __launch_bounds__(256)
void ctx_wmma_kernel(const float* __restrict__ IN, const float* __restrict__ w,
                     float* __restrict__ c_part) {
    __shared__ float wl[2048];
    const int chunk = blockIdx.x >> 5;      // 4 chunks of 2048 seq positions
    const int kb    = chunk * 2048;
    for (int i = threadIdx.x; i < 2048; i += 256) wl[i] = w[kb + i];
    __syncthreads();

    const int lane = threadIdx.x & 31;
    const int wave = threadIdx.x >> 5;
    const int tile = (blockIdx.x & 31) * 8 + wave;   // 0..255 -> 16-col d tile
    const int d0   = tile * 16;
    const int half = (lane >> 4) << 1;               // 0 for lanes 0-15, 2 for 16-31
    const float sel = ((lane & 15) == 0) ? 1.0f : 0.0f;

    const float*  bptr = IN + (size_t)(kb + half) * DIMS + d0 + (lane & 15);
    const float2* wp   = (const float2*)(wl + half);   // wp[k/2] = wl[k+half..+1]

    v8f acc0 = {};
    v8f acc1 = {};
    for (int k = 0; k < 2048; k += 8) {
        // --- K-step k..k+3 ---
        float2 wv0 = wp[k >> 1];
        v2f a0; a0.x = wv0.x * sel; a0.y = wv0.y * sel;
        const float* p0 = bptr + (size_t)k * DIMS;
        v2f b0; b0.x = p0[0]; b0.y = p0[DIMS];
        __builtin_prefetch(p0 + 8 * DIMS, 0, 1);
        acc0 = __builtin_amdgcn_wmma_f32_16x16x4_f32(
            false, a0, false, b0, (short)0, acc0, false, false);

        // --- K-step k+4..k+7 (independent accumulator) ---
        float2 wv1 = wp[(k + 4) >> 1];
        v2f a1; a1.x = wv1.x * sel; a1.y = wv1.y * sel;
        const float* p1 = bptr + (size_t)(k + 4) * DIMS;
        v2f b1; b1.x = p1[0]; b1.y = p1[DIMS];
        acc1 = __builtin_amdgcn_wmma_f32_16x16x4_f32(
            false, a1, false, b1, (short)0, acc1, false, false);
    }
    if (lane < 16) c_part[(size_t)chunk * DIMS + d0 + lane] = acc0[0] + acc1[0];
}

// ---------------------------------------------------------------------------
extern "C" void kernel_launch(void* const* d_in, const int* in_sizes, int n_in,
                              void* d_out, int out_size, void* d_ws, size_t ws_size,
                              hipStream_t stream) {
    (void)in_sizes; (void)n_in; (void)out_size; (void)ws_size;
    const float* prev = (const float*)d_in[0];   // (8191, 4096)
    const float* inp  = (const float*)d_in[1];   // (4096,)
    const float* WQ   = (const float*)d_in[2];   // (4096, 4096)
    const float* WK   = (const float*)d_in[3];
    const float* WV   = (const float*)d_in[4];

    float* out        = (float*)d_out;           // [output(4096) | inputs(8192*4096)]
    float* out_inputs = out + DIMS;

    float* ws      = (float*)d_ws;
    float* q       = ws;                  // 4096
    float* kq      = ws + 4096;           // 4096
    float* logits  = ws + 8192;           // 8192
    float* weights = ws + 16384;          // 8192
    float* cvec    = ws + 24576;          // 4096
    float* kq_part = ws + 28672;          // 8 * 4096
    float* c_part  = kq_part + 8 * DIMS;  // 4 * 4096   (total ~304 KB)

    // q = W_Q @ inp
    gemv_rows_kernel<<<512, 256, 0, stream>>>(WQ, inp, nullptr, q);
    // kq = W_K^T @ q
    kq_partial_kernel<<<128, 256, 0, stream>>>(WK, q, kq_part);
    reduce_part_kernel<<<16, 256, 0, stream>>>(kq_part, kq, 8);
    // inputs copy + logits = inputs @ kq
    logits_copy_kernel<<<NROWS, 256, 0, stream>>>(prev, inp, kq, out_inputs, logits);
    // weights = softmax(logits)
    softmax_kernel<<<1, 256, 0, stream>>>(logits, weights);
    // c = weights @ inputs   (WMMA f32 16x16x4)
    ctx_wmma_kernel<<<128, 256, 0, stream>>>(out_inputs, weights, c_part);
    reduce_part_kernel<<<16, 256, 0, stream>>>(c_part, cvec, 4);
    // output = W_V @ c + inp
    gemv_rows_kernel<<<512, 256, 0, stream>>>(WV, cvec, inp, out);
}